// GATLayer_2_35424890258181
// MI455X (gfx1250) — compile-verified
//
#include <hip/hip_runtime.h>
#include <hip/hip_bf16.h>

typedef __attribute__((ext_vector_type(2))) float v2f;
typedef __attribute__((ext_vector_type(8))) float v8f;

#define GAT_N 100000
#define GAT_E 1600000
#define GAT_INF 128
#define GAT_H 8
#define GAT_F 8
#define GAT_HF 64
#define NEG_SLOPE 0.2f
#define MTILES (GAT_N / 16)   // 6250, exact

// ---------------------------------------------------------------------------
// K0: proj = x @ W via v_wmma_f32_16x16x4_f32.
// One wave (32 lanes) computes a 16x64 output tile: 4 n-tiles of 16 cols,
// K=128 -> 32 k-steps -> 128 WMMA issues per wave.
// A frag (16x4 f32): lane L holds m = L%16, k = k0 + (L/16)*2 + {0,1}
// B frag (4x16 f32): lane L holds n = L%16, k = k0 + (L/16)*2 + {0,1}
// C/D (16x16 f32):  vgpr v, lane L -> m = v + 8*(L/16), n = L%16
// ---------------------------------------------------------------------------
__global__ __launch_bounds__(256) void gat_proj_wmma(
    const float* __restrict__ x, const float* __restrict__ W,
    float* __restrict__ proj) {
  const int lane = threadIdx.x & 31;
  const int wave = threadIdx.x >> 5;
  const int mtile = blockIdx.x * 8 + wave;
  if (mtile >= MTILES) return;  // wave-uniform exit: EXEC all-1s for WMMA
  const int m0 = mtile * 16;
  const int lm = lane & 15;         // m (for A) or n (for B/C)
  const int lk = (lane >> 4) * 2;   // k sub-offset 0 or 2

  v8f acc0 = {}, acc1 = {}, acc2 = {}, acc3 = {};
  const float* xrow = x + (size_t)(m0 + lm) * GAT_INF;

#pragma unroll
  for (int k0 = 0; k0 < GAT_INF; k0 += 4) {
    v2f a;
    a.x = xrow[k0 + lk];
    a.y = xrow[k0 + lk + 1];
    const float* wp = W + (size_t)(k0 + lk) * GAT_HF + lm;
    v2f b0, b1, b2, b3;
    b0.x = wp[0];       b0.y = wp[GAT_HF];
    b1.x = wp[16];      b1.y = wp[GAT_HF + 16];
    b2.x = wp[32];      b2.y = wp[GAT_HF + 32];
    b3.x = wp[48];      b3.y = wp[GAT_HF + 48];
    acc0 = __builtin_amdgcn_wmma_f32_16x16x4_f32(false, a, false, b0, (short)0, acc0, false, false);
    acc1 = __builtin_amdgcn_wmma_f32_16x16x4_f32(false, a, false, b1, (short)0, acc1, false, false);
    acc2 = __builtin_amdgcn_wmma_f32_16x16x4_f32(false, a, false, b2, (short)0, acc2, false, false);
    acc3 = __builtin_amdgcn_wmma_f32_16x16x4_f32(false, a, false, b3, (short)0, acc3, false, false);
  }

  const int mrow = (lane >> 4) * 8;
  float* prow = proj + (size_t)(m0 + mrow) * GAT_HF + lm;
#pragma unroll
  for (int v = 0; v < 8; ++v) {
    prow[(size_t)v * GAT_HF + 0]  = acc0[v];
    prow[(size_t)v * GAT_HF + 16] = acc1[v];
    prow[(size_t)v * GAT_HF + 32] = acc2[v];
    prow[(size_t)v * GAT_HF + 48] = acc3[v];
  }
}

// ---------------------------------------------------------------------------
// K1: per-(node, head) attention scalars; also zero denom and the output
// accumulator (d_out), since the harness poisons them.
// ---------------------------------------------------------------------------
__global__ __launch_bounds__(256) void gat_scores(
    const float* __restrict__ proj, const float* __restrict__ a_self,
    const float* __restrict__ a_neigh, float* __restrict__ s_self,
    float* __restrict__ s_neigh, float* __restrict__ denom,
    float* __restrict__ out_accum) {
  const int i = blockIdx.x * blockDim.x + threadIdx.x;
  if (i >= GAT_N * GAT_H) return;
  const int n = i >> 3;
  const int h = i & 7;
  const float* p = proj + (size_t)n * GAT_HF + h * GAT_F;
  const float* as = a_self + h * GAT_F;
  const float* an = a_neigh + h * GAT_F;
  float ss = 0.f, sn = 0.f;
#pragma unroll
  for (int f = 0; f < GAT_F; ++f) {
    const float v = p[f];
    ss += v * as[f];
    sn += v * an[f];
  }
  s_self[i] = ss;
  s_neigh[i] = sn;
  denom[i] = 0.f;
  float4* oz = (float4*)(out_accum + (size_t)n * GAT_HF + h * GAT_F);
  oz[0] = make_float4(0.f, 0.f, 0.f, 0.f);
  oz[1] = make_float4(0.f, 0.f, 0.f, 0.f);
}

// ---------------------------------------------------------------------------
// K2: per-edge softmax denominators: denom[tgt,h] += exp(leaky_relu(...))
// ---------------------------------------------------------------------------
__global__ __launch_bounds__(256) void gat_denom(
    const long long* __restrict__ ei, const float* __restrict__ s_self,
    const float* __restrict__ s_neigh, float* __restrict__ denom) {
  const int e = blockIdx.x * blockDim.x + threadIdx.x;
  if (e >= GAT_E) return;
  const long long src = ei[e];
  const long long tgt = ei[GAT_E + e];
  const float* ps = s_self + (size_t)src * GAT_H;
  const float* pn = s_neigh + (size_t)tgt * GAT_H;
  float* pd = denom + (size_t)tgt * GAT_H;
#pragma unroll
  for (int h = 0; h < GAT_H; ++h) {
    float sc = ps[h] + pn[h];
    sc = sc > 0.f ? sc : NEG_SLOPE * sc;
    unsafeAtomicAdd(pd + h, __expf(sc));
  }
}

// ---------------------------------------------------------------------------
// K3: per-(edge, head): alpha = exp(score)/denom[tgt,h]; accumulate
// alpha * proj[src, h, :] into out[tgt, h, :] via hardware fadd atomics.
// ---------------------------------------------------------------------------
__global__ __launch_bounds__(256) void gat_aggregate(
    const long long* __restrict__ ei, const float* __restrict__ s_self,
    const float* __restrict__ s_neigh, const float* __restrict__ denom,
    const float* __restrict__ proj, float* __restrict__ out) {
  const int idx = blockIdx.x * blockDim.x + threadIdx.x;
  if (idx >= GAT_E * GAT_H) return;
  const int e = idx >> 3;
  const int h = idx & 7;
  const long long src = ei[e];
  const long long tgt = ei[GAT_E + e];
  float sc = s_self[(size_t)src * GAT_H + h] + s_neigh[(size_t)tgt * GAT_H + h];
  sc = sc > 0.f ? sc : NEG_SLOPE * sc;
  const float alpha = __expf(sc) / denom[(size_t)tgt * GAT_H + h];
  const float* p = proj + (size_t)src * GAT_HF + h * GAT_F;
  float* o = out + (size_t)tgt * GAT_HF + h * GAT_F;
#pragma unroll
  for (int f = 0; f < GAT_F; ++f) {
    unsafeAtomicAdd(o + f, p[f] * alpha);
  }
}

// ---------------------------------------------------------------------------
// K4: in-place ELU on the accumulated output.
// ---------------------------------------------------------------------------
__global__ __launch_bounds__(256) void gat_elu(float* __restrict__ out, int n) {
  const int i = blockIdx.x * blockDim.x + threadIdx.x;
  if (i >= n) return;
  const float v = out[i];
  out[i] = v > 0.f ? v : expm1f(v);
}

extern "C" void kernel_launch(void* const* d_in, const int* in_sizes, int n_in,
                              void* d_out, int out_size, void* d_ws, size_t ws_size,
                              hipStream_t stream) {
  const float* x = (const float*)d_in[0];
  const long long* ei = (const long long*)d_in[1];
  const float* W = (const float*)d_in[2];
  const float* a_self = (const float*)d_in[3];
  const float* a_neigh = (const float*)d_in[4];
  float* out = (float*)d_out;

  float* ws = (float*)d_ws;
  float* proj = ws;                               // N*64
  float* s_self = proj + (size_t)GAT_N * GAT_HF;  // N*8
  float* s_neigh = s_self + (size_t)GAT_N * GAT_H;
  float* denom = s_neigh + (size_t)GAT_N * GAT_H;

  // K0: WMMA projection (one wave per 16-row tile, 8 waves per block)
  gat_proj_wmma<<<(MTILES + 7) / 8, 256, 0, stream>>>(x, W, proj);
  // K1: per-(node,head) scalars + zero denom + zero output accumulator
  gat_scores<<<(GAT_N * GAT_H + 255) / 256, 256, 0, stream>>>(
      proj, a_self, a_neigh, s_self, s_neigh, denom, out);
  // K2: softmax denominators
  gat_denom<<<(GAT_E + 255) / 256, 256, 0, stream>>>(ei, s_self, s_neigh, denom);
  // K3: weighted scatter-add
  gat_aggregate<<<(GAT_E * GAT_H + 255) / 256, 256, 0, stream>>>(
      ei, s_self, s_neigh, denom, proj, out);
  // K4: ELU
  gat_elu<<<(GAT_N * GAT_HF + 255) / 256, 256, 0, stream>>>(out, GAT_N * GAT_HF);
}